// NeRFEmissionAbsorptionRaymarcher_84507776516823
// MI455X (gfx1250) — compile-verified
//
#include <hip/hip_runtime.h>
#include <stdint.h>

// ---------------------------------------------------------------------------
// NeRF emission-absorption raymarcher for MI455X (gfx1250, wave32).
//
// Shapes (from reference): B=8, R=16384, N=128, F=3.
//   rays_densities (B,R,N,1) f32   64 MiB
//   rays_features  (B,R,N,3) f32  192 MiB
//   lengths        (B,R,N)   f32   64 MiB
//   out            (B,R,4)   f32    2 MiB
// 320 MiB single-pass @ 23.3 TB/s  =>  ~14 us floor. Memory bound; the job is
// coalesced NT streaming + hiding the per-ray serial scan. One wave per ray,
// 4 samples per lane. Features staged via CDNA5 async global->LDS copies
// (ASYNCcnt / s_wait_asynccnt), double buffered across grid-stride iterations.
// ---------------------------------------------------------------------------

#define WPB            8                 // waves per block (256 threads)
#define BLOCK_THREADS  (WPB * 32)
#define N_SAMP         128
#define FEAT_FLOATS    (N_SAMP * 3)      // 384 floats (1536 B) per ray
#define RAYS_PER_WAVE  4                 // grid-stride depth (enables pipelining)

typedef float v4f __attribute__((ext_vector_type(4)));

// Types matching the async-copy builtin's signature:
//   param0: int __vector(4) addrspace(1)*   (global, printed '__device__')
//   param1: int __vector(4) addrspace(3)*   (LDS,    printed '__shared__')
typedef int v4i __attribute__((vector_size(4 * sizeof(int))));
typedef __attribute__((address_space(1))) v4i gv4i_t;   // global
typedef __attribute__((address_space(3))) v4i sv4i_t;   // LDS

// Issue 3 coalesced async B128 copies: 32 lanes x 16B x 3 = 1536 B = one ray's
// features, global -> LDS, bypassing VGPRs. IOFFSET is added to BOTH the
// global and LDS addresses (CDNA5 ISA), so a single per-lane
// base covers all three transfers.
__device__ __forceinline__ void async_issue_features(const float* gbase,
                                                     float* lbase, int lane) {
  // generic->AS casts via integer: global flat addr == AS1 addr; LDS offset
  // lives in flat addr[31:0] (CDNA5 ISA aperture mapping).
  gv4i_t* g = (gv4i_t*)(uintptr_t)(gbase + lane * 4);
  sv4i_t* l = (sv4i_t*)(uint32_t)(uintptr_t)(lbase + lane * 4);
#if __has_builtin(__builtin_amdgcn_global_load_async_to_lds_b128)
  __builtin_amdgcn_global_load_async_to_lds_b128(g, l, 0, 0);
  __builtin_amdgcn_global_load_async_to_lds_b128(g, l, 512, 0);
  __builtin_amdgcn_global_load_async_to_lds_b128(g, l, 1024, 0);
#else
  asm volatile(
      "global_load_async_to_lds_b128 %0, %1, off\n\t"
      "global_load_async_to_lds_b128 %0, %1, off offset:512\n\t"
      "global_load_async_to_lds_b128 %0, %1, off offset:1024"
      :: "v"((uint32_t)(uintptr_t)l), "v"((uint64_t)(uintptr_t)g)
      : "memory");
#endif
}

// ASYNCcnt waits: async->LDS loads complete in order, so waiting for <=3
// outstanding guarantees the *current* ray's 3 copies landed while the next
// ray's 3 stay in flight (double buffer).
__device__ __forceinline__ void wait_async_le(int which /*0 or 3, constant*/) {
#if __has_builtin(__builtin_amdgcn_s_wait_asynccnt)
  if (which == 3) __builtin_amdgcn_s_wait_asynccnt(3);
  else            __builtin_amdgcn_s_wait_asynccnt(0);
#else
  if (which == 3) asm volatile("s_wait_asynccnt 0x3" ::: "memory");
  else            asm volatile("s_wait_asynccnt 0x0" ::: "memory");
#endif
  asm volatile("" ::: "memory");  // keep LDS reads below the wait
}

__global__ __launch_bounds__(BLOCK_THREADS) void
NeRF_EA_raymarch_kernel(const float* __restrict__ dens,   // (BR,128)
                        const float* __restrict__ feat,   // (BR,128,3)
                        const float* __restrict__ len,    // (BR,128)
                        float* __restrict__ out,          // (BR,4)
                        int n_rays) {
  __shared__ __align__(16) float sfeat[2][WPB][FEAT_FLOATS];  // 24 KiB

  const int lane  = threadIdx.x & 31;
  const int wslot = threadIdx.x >> 5;

  const int wave0   = blockIdx.x * WPB + wslot;   // uniform per wave
  const int wstride = gridDim.x * WPB;
  if (wave0 >= n_rays) return;                    // whole-wave uniform exit

  // Prologue: stage first ray's features into buffer 0.
  async_issue_features(feat + (size_t)wave0 * FEAT_FLOATS, sfeat[0][wslot], lane);

  int buf = 0;
  for (int ray = wave0; ray < n_rays; ray += wstride, buf ^= 1) {
    const int next = ray + wstride;               // uniform per wave
    if (next < n_rays) {
      // Pipeline: next ray's features -> other LDS buffer; prefetch its
      // density/length cachelines (global_prefetch_b8).
      async_issue_features(feat + (size_t)next * FEAT_FLOATS,
                           sfeat[buf ^ 1][wslot], lane);
      __builtin_prefetch(dens + (size_t)next * N_SAMP + lane * 4, 0, 0);
      __builtin_prefetch(len  + (size_t)next * N_SAMP + lane * 4, 0, 0);
    }

    // Densities & lengths: 16B/lane fully coalesced, non-temporal (one-pass
    // 320 MiB stream > 192 MB L2 -> don't rinse the cache).
    const v4f dv = __builtin_nontemporal_load(
        (const v4f*)(dens + (size_t)ray * N_SAMP + lane * 4));
    const v4f lv = __builtin_nontemporal_load(
        (const v4f*)(len + (size_t)ray * N_SAMP + lane * 4));

    // Transmittance t = 1 + 1e-10 - d ; (1.0f + 1e-10f) == 1.0f exactly in
    // f32, matching the reference's float32 arithmetic.
    const float t0 = 1.0f - dv.x, t1 = 1.0f - dv.y;
    const float t2 = 1.0f - dv.z, t3 = 1.0f - dv.w;
    const float e1 = t0, e2 = t0 * t1, e3 = e2 * t2;  // lane-local exclusive
    float p = e3 * t3;                                // lane's total product

    // Inclusive prefix product across 32 lanes (5 shfl_up steps).
#pragma unroll
    for (int delta = 1; delta < 32; delta <<= 1) {
      const float o = __shfl_up(p, (unsigned)delta, 32);
      if (lane >= delta) p *= o;
    }
    // Exclusive prefix for this lane's first sample.
    float E = __shfl_up(p, 1u, 32);
    if (lane == 0) E = 1.0f;

    // weights[i] = d[i] * prod_{j<i} t[j]
    const float w0 = dv.x * E;
    const float w1 = dv.y * (E * e1);
    const float w2 = dv.z * (E * e2);
    const float w3 = dv.w * (E * e3);

    float alpha = (w0 + w1) + (w2 + w3);
    float depth = w0 * lv.x + w1 * lv.y + w2 * lv.z + w3 * lv.w;
    const float last_len = __shfl(lv.w, 31, 32);   // lengths[..., -1]

    // Consume this ray's staged features (next ray's 3 copies may remain).
    wait_async_le(next < n_rays ? 3 : 0);

    // Lane's 4 samples x 3 channels = 12 floats at 48B-aligned LDS offset.
    const float* swf = &sfeat[buf][wslot][lane * 12];
    const v4f f0 = *(const v4f*)(swf + 0);  // s0.xyz s1.x
    const v4f f1 = *(const v4f*)(swf + 4);  // s1.yz  s2.xy
    const v4f f2 = *(const v4f*)(swf + 8);  // s2.z   s3.xyz

    float cr = w0 * f0.x + w1 * f0.w + w2 * f1.z + w3 * f2.y;
    float cg = w0 * f0.y + w1 * f1.x + w2 * f1.w + w3 * f2.z;
    float cb = w0 * f0.z + w1 * f1.y + w2 * f2.x + w3 * f2.w;

    // Butterfly-sum the 5 accumulators across the wave.
#pragma unroll
    for (int m = 16; m >= 1; m >>= 1) {
      cr    += __shfl_xor(cr, m, 32);
      cg    += __shfl_xor(cg, m, 32);
      cb    += __shfl_xor(cb, m, 32);
      depth += __shfl_xor(depth, m, 32);
      alpha += __shfl_xor(alpha, m, 32);
    }

    if (lane == 0) {
      const float bgw = 1.0f - alpha;              // white background weight
      v4f o;
      o.x = cr + bgw;
      o.y = cg + bgw;
      o.z = cb + bgw;
      o.w = depth + bgw * last_len;
      *(v4f*)(out + (size_t)ray * 4) = o;
    }
  }
}

extern "C" void kernel_launch(void* const* d_in, const int* in_sizes, int n_in,
                              void* d_out, int out_size, void* d_ws,
                              size_t ws_size, hipStream_t stream) {
  (void)n_in; (void)d_ws; (void)ws_size; (void)out_size;
  const float* dens = (const float*)d_in[0];   // (B,R,N,1) f32
  const float* feat = (const float*)d_in[1];   // (B,R,N,3) f32
  const float* lens = (const float*)d_in[2];   // (B,R,N)   f32
  float* out = (float*)d_out;                  // (B,R,4)   f32

  const int n_rays = in_sizes[0] / N_SAMP;     // B*R = 131072
  const int waves  = (n_rays + RAYS_PER_WAVE - 1) / RAYS_PER_WAVE;
  const int blocks = (waves + WPB - 1) / WPB;  // 4096 blocks of 8 waves

  NeRF_EA_raymarch_kernel<<<blocks, BLOCK_THREADS, 0, stream>>>(
      dens, feat, lens, out, n_rays);
}